// similarity_model_79748952752687
// MI455X (gfx1250) — compile-verified
//
#include <hip/hip_runtime.h>
#include <hip/hip_bf16.h>

#define BATCH    1024
#define VOCAB    100000
#define EMB      300
#define KT       (EMB / 4)        // 75 k-steps of 4
#define NT_TOTAL (VOCAB / 16)     // 6250 column tiles
#define MROWS    32               // batch rows per workgroup (2 WMMA M-tiles)
#define KEEP     11               // top-(k+1)
#define TOPK     10
#define MAX_NCHUNK 64

typedef __attribute__((ext_vector_type(2))) float v2f;
typedef __attribute__((ext_vector_type(8))) float v8f;

// Sorted insertion into a descending top-KEEP register list.
// Arrays stay in registers (constant indices after full unroll).
__device__ __forceinline__ void topk_insert(float* tv, int* ti, float cv, int ci) {
#pragma unroll
    for (int i = 0; i < KEEP; ++i) {
        if (cv > tv[i]) {
            float fv = tv[i]; int fi = ti[i];
            tv[i] = cv; ti[i] = ci;
            cv = fv; ci = fi;
        }
    }
}

// One wave (32 lanes) per workgroup. blockIdx.x = batch tile (32 rows),
// blockIdx.y = vocab chunk. Each B fragment feeds two independent WMMAs.
// Each lane owns exactly one batch row's top-11 list (no merge stage).
__global__ __launch_bounds__(32) void gemm_topk_kernel(
    const int* __restrict__ wordid,
    const float* __restrict__ weight,
    float* __restrict__ ws_val,     // [BATCH][nchunk][KEEP]
    int* __restrict__ ws_idx,       // [BATCH][nchunk][KEEP]
    int tiles_per_chunk)
{
    __shared__ float lds_a[MROWS][302];   // A tile, padded stride (8B-aligned rows)
    __shared__ float lds_c[MROWS][17];    // C spill for row-major scan

    const int lane  = threadIdx.x;
    const int m0    = blockIdx.x * MROWS;
    const int chunk = blockIdx.y;
    const int nchunk = gridDim.y;

    // ---- gather A tile: 32 wordvec rows into LDS (uniform 300 iters/lane) ----
    for (int idx = lane; idx < MROWS * EMB; idx += 32) {
        int r = idx / EMB;
        int k = idx - r * EMB;
        int row = wordid[m0 + r];
        lds_a[r][k] = weight[(size_t)row * EMB + k];
    }
    __syncthreads();

    const int r = lane & 15;   // WMMA fragment column / row-within-tile
    const int h = lane >> 4;   // half selector (K pair / M+8)

    float tv[KEEP]; int ti[KEEP];
#pragma unroll
    for (int i = 0; i < KEEP; ++i) { tv[i] = -3.0e38f; ti[i] = -1; }

    const int t0 = chunk * tiles_per_chunk;
    int t1 = t0 + tiles_per_chunk;
    if (t1 > NT_TOTAL) t1 = NT_TOTAL;

    for (int t = t0; t < t1; ++t) {
        const int n0 = t * 16;
        // B fragment source: lane supplies B[K=2h+j][N=r] = W[n0+r][4kt+2h+j]
        const float* wp = weight + (size_t)(n0 + r) * EMB + 2 * h;
        if (t + 1 < t1) __builtin_prefetch(wp + (size_t)16 * EMB, 0, 1);

        v8f acc0 = {0.0f, 0.0f, 0.0f, 0.0f, 0.0f, 0.0f, 0.0f, 0.0f};
        v8f acc1 = {0.0f, 0.0f, 0.0f, 0.0f, 0.0f, 0.0f, 0.0f, 0.0f};
#pragma unroll 5
        for (int kt = 0; kt < KT; ++kt) {
            v2f b  = *(const v2f*)(wp + 4 * kt);                      // 8B aligned
            v2f a0 = *(const v2f*)(&lds_a[r][4 * kt + 2 * h]);        // M-tile 0
            v2f a1 = *(const v2f*)(&lds_a[16 + r][4 * kt + 2 * h]);   // M-tile 1
            acc0 = __builtin_amdgcn_wmma_f32_16x16x4_f32(
                false, a0, false, b, (short)0, acc0, false, false);
            acc1 = __builtin_amdgcn_wmma_f32_16x16x4_f32(
                false, a1, false, b, (short)0, acc1, false, false);
        }

        // spill C: lane holds col N=r, rows M = g + 8h (+16 for tile 1)
        __syncthreads();   // protect WAR vs previous tile's scan
#pragma unroll
        for (int g = 0; g < 8; ++g) {
            lds_c[g + 8 * h][r]      = acc0[g];
            lds_c[16 + g + 8 * h][r] = acc1[g];
        }
        __syncthreads();

        // scan: lane L owns batch row m0+L, reads its full 16 sims
#pragma unroll
        for (int c = 0; c < 16; ++c) {
            float v = lds_c[lane][c];
            if (v > tv[KEEP - 1]) topk_insert(tv, ti, v, n0 + c);
        }
    }

    // ---- write this lane's partial top-11 ----
    float* ov = ws_val + ((size_t)(m0 + lane) * nchunk + chunk) * KEEP;
    int*   oi = ws_idx + ((size_t)(m0 + lane) * nchunk + chunk) * KEEP;
#pragma unroll
    for (int i = 0; i < KEEP; ++i) { ov[i] = tv[i]; oi[i] = ti[i]; }
}

// One thread per batch row: merge nchunk partial lists, emit top 1..10
// (drop column 0 = the self-similarity max, matching score[:,1:]).
__global__ __launch_bounds__(256) void final_topk_kernel(
    const float* __restrict__ ws_val,
    const int* __restrict__ ws_idx,
    float* __restrict__ out,
    int nchunk)
{
    int b = blockIdx.x * blockDim.x + threadIdx.x;
    if (b >= BATCH) return;

    float tv[KEEP]; int ti[KEEP];
#pragma unroll
    for (int i = 0; i < KEEP; ++i) { tv[i] = -3.0e38f; ti[i] = -1; }

    const float* v  = ws_val + (size_t)b * nchunk * KEEP;
    const int*   ix = ws_idx + (size_t)b * nchunk * KEEP;
    const int n = nchunk * KEEP;
    for (int j = 0; j < n; ++j) {
        float cv = v[j];
        if (cv > tv[KEEP - 1]) topk_insert(tv, ti, cv, ix[j]);
    }

#pragma unroll
    for (int j = 1; j < KEEP; ++j) {
        out[(size_t)b * TOPK + (j - 1)] = tv[j];
        out[(size_t)BATCH * TOPK + (size_t)b * TOPK + (j - 1)] = (float)ti[j];
    }
}

extern "C" void kernel_launch(void* const* d_in, const int* in_sizes, int n_in,
                              void* d_out, int out_size, void* d_ws, size_t ws_size,
                              hipStream_t stream) {
    const int*   wordid = (const int*)d_in[0];
    const float* weight = (const float*)d_in[1];
    float*       out    = (float*)d_out;

    // size vocab chunking to available scratch (cap at MAX_NCHUNK chunks)
    const size_t per_chunk = (size_t)BATCH * KEEP * (sizeof(float) + sizeof(int));
    int nchunk = (ws_size >= per_chunk) ? (int)(ws_size / per_chunk) : 1;
    if (nchunk > MAX_NCHUNK) nchunk = MAX_NCHUNK;
    if (nchunk < 1) nchunk = 1;
    const int tiles_per_chunk = (NT_TOTAL + nchunk - 1) / nchunk;

    float* ws_val = (float*)d_ws;
    int*   ws_idx = (int*)((char*)d_ws + (size_t)BATCH * nchunk * KEEP * sizeof(float));

    dim3 grid(BATCH / MROWS, nchunk);
    gemm_topk_kernel<<<grid, 32, 0, stream>>>(wordid, weight, ws_val, ws_idx,
                                              tiles_per_chunk);
    final_topk_kernel<<<(BATCH + 255) / 256, 256, 0, stream>>>(ws_val, ws_idx,
                                                               out, nchunk);
}